// BendingLoss_58334245814720
// MI455X (gfx1250) — compile-verified
//
#include <hip/hip_runtime.h>
#include <stdint.h>

// BendingLoss on MI455X (gfx1250, wave32).
// Phase 1: 32 MB of target channel-1 -> per-image 8KB contour-source bitmasks
//          via coalesced loads + wave32 ballot (32:1 compression at HBM rate).
// Phase 2: one block per image; TDM (tensor_load_to_lds) pulls the 8KB bitmask
//          into LDS, bit-parallel 3x3 stencil builds contour bits, prev/next
//          contour points found by ctz/clz bit scans, per-image energy reduced.
// Phase 3: deterministic serial sum of 128 partials * (1/B).

#define H_    256
#define W_    256
#define PIX   (H_ * W_)          // 65536 pixels / image
#define WORDS (PIX / 32)         // 2048 u32 bitmask words / image
#define MU_F  0.75f

typedef unsigned int uint4v __attribute__((ext_vector_type(4)));
typedef int          int4v  __attribute__((ext_vector_type(4)));
typedef int          int8v  __attribute__((ext_vector_type(8)));

// ---------------- Phase 1: threshold + ballot-compress to bitmask ----------
__global__ __launch_bounds__(256)
void mask_bits_kernel(const float* __restrict__ target,
                      unsigned* __restrict__ bits, int nWaves) {
  const int lane = threadIdx.x & 31;
  const int wave = blockIdx.x * (blockDim.x >> 5) + (threadIdx.x >> 5);
  if (wave >= nWaves) return;               // wave-uniform exit
  const int b  = wave >> 8;                 // 256 waves per image (2048/8 words)
  const int w0 = (wave & 255) * 8;          // this wave's first of 8 words
  const float* plane = target + (size_t)b * (2 * PIX) + PIX;  // channel 1
  float v[8];
#pragma unroll
  for (int j = 0; j < 8; ++j)               // 8 independent 128B-coalesced loads
    v[j] = plane[(w0 + j) * 32 + lane];
  unsigned r[8];
#pragma unroll
  for (int j = 0; j < 8; ++j)               // wave32 ballot: 32 pixels -> 1 word
    r[j] = (unsigned)__ballot(v[j] > 0.5f);
  if (lane == 0) {
    uint4* dst = (uint4*)(bits + (size_t)b * WORDS + w0);
    dst[0] = make_uint4(r[0], r[1], r[2], r[3]);
    dst[1] = make_uint4(r[4], r[5], r[6], r[7]);
  }
}

// ---------------- Phase 2: contour + bending energy, one block per image ---
__global__ __launch_bounds__(256)
void contour_energy_kernel(const unsigned* __restrict__ bits,
                           float* __restrict__ perImage) {
  __shared__ unsigned mw[WORDS];   // mask bitmask (8 KB)
  __shared__ unsigned cw[WORDS];   // contour bitmask (8 KB)
  __shared__ int sFirst, sLast;
  __shared__ float sPart[8];
  const int b = blockIdx.x;
  const unsigned* src = bits + (size_t)b * WORDS;

  if (threadIdx.x == 0) { sFirst = PIX; sLast = -1; }

#if __has_builtin(__builtin_amdgcn_tensor_load_to_lds) && \
    __has_builtin(__builtin_amdgcn_s_wait_tensorcnt)
  // CDNA5 Tensor Data Mover: DMA the 8KB tile global->LDS (wave 0 issues).
  if (threadIdx.x < 32) {
    const unsigned long long ga = (unsigned long long)(uintptr_t)src;
    const unsigned la = (unsigned)(unsigned long long)(uintptr_t)&mw[0];
    uint4v g0;
    g0.x = 1u;                                   // count=1 (valid), user mode
    g0.y = la;                                   // lds_addr (bytes)
    g0.z = (unsigned)(ga & 0xFFFFFFFFull);       // global_addr[31:0]
    g0.w = (unsigned)((ga >> 32) & 0x01FFFFFFull) | (2u << 30); // addr hi|type=2
    int8v g1;
    g1[0] = (int)(2u << 16);                     // data_size=4B; mask/pad/iter=0
    g1[1] = (int)((WORDS & 0xFFFF) << 16);       // tensor_dim0 lo16 (=2048)
    g1[2] = (int)((WORDS >> 16) | (1u << 16));   // tensor_dim0 hi | tensor_dim1=1
    g1[3] = (int)((unsigned)WORDS << 16);        // tile_dim0 = 2048
    g1[4] = 1;                                   // tile_dim1=1, tile_dim2=0
    g1[5] = WORDS;                               // tensor_dim0_stride lo32
    g1[6] = (int)((unsigned)WORDS << 16);        // stride0 hi | stride1 lo16
    g1[7] = 0;
    int4v z4 = {0, 0, 0, 0};
#if defined(__clang_major__) && (__clang_major__ >= 23)
    int8v z8 = {0, 0, 0, 0, 0, 0, 0, 0};
    __builtin_amdgcn_tensor_load_to_lds(g0, g1, z4, z4, z8, 0);
#else
    __builtin_amdgcn_tensor_load_to_lds(g0, g1, z4, z4, 0);
#endif
    __builtin_amdgcn_s_wait_tensorcnt(0);        // TENSORcnt==0 before use
  }
#else
  for (int i = threadIdx.x; i < WORDS; i += blockDim.x) mw[i] = src[i];
#endif
  __syncthreads();

  // Bit-parallel 3x3 stencil: contour = mask & ~(all 8 neighbors set).
  for (int i = threadIdx.x; i < WORDS; i += blockDim.x) {
    const int r = i >> 3, wc = i & 7;
    const unsigned Bc = mw[i];
    const unsigned Bl = wc ? mw[i - 1] : 0u;
    const unsigned Br = (wc < 7) ? mw[i + 1] : 0u;
    unsigned Ac = 0u, Al = 0u, Ar = 0u, Cc = 0u, Cl = 0u, Cr = 0u;
    if (r > 0)      { Ac = mw[i - 8]; Al = wc ? mw[i - 9] : 0u; Ar = (wc < 7) ? mw[i - 7] : 0u; }
    if (r < H_ - 1) { Cc = mw[i + 8]; Cl = wc ? mw[i + 7] : 0u; Cr = (wc < 7) ? mw[i + 9] : 0u; }
    const unsigned all8 =
        ((Ac << 1) | (Al >> 31)) & Ac & ((Ac >> 1) | (Ar << 31)) &
        ((Bc << 1) | (Bl >> 31))      & ((Bc >> 1) | (Br << 31)) &
        ((Cc << 1) | (Cl >> 31)) & Cc & ((Cc >> 1) | (Cr << 31));
    const unsigned c = Bc & ~all8;
    cw[i] = c;
    if (c) {                                    // track global first/last point
      atomicMin(&sFirst, (i << 5) + __builtin_ctz(c));
      atomicMax(&sLast,  (i << 5) + 31 - __builtin_clz(c));
    }
  }
  __syncthreads();
  const int first = sFirst, last = sLast;

  // Per contour point (except global first/last): energy from prev/next point
  // in row-major order == nearest set bits below/above in the bitmask.
  float acc = 0.f;
  for (int i = threadIdx.x; i < WORDS; i += blockDim.x) {
    const unsigned cwi = cw[i];
    unsigned rem = cwi;
    while (rem) {
      const int k = __builtin_ctz(rem);
      rem &= rem - 1u;                          // rem now == bits above k
      const int p = (i << 5) + k;
      if (p == first || p == last) continue;
      // previous contour point
      const unsigned below = (k == 0) ? 0u : (cwi & ((1u << k) - 1u));
      int pp;
      if (below) pp = (i << 5) + 31 - __builtin_clz(below);
      else { int j = i - 1; while (cw[j] == 0u) --j;
             pp = (j << 5) + 31 - __builtin_clz(cw[j]); }
      // next contour point
      int pn;
      if (rem) pn = (i << 5) + __builtin_ctz(rem);
      else { int j = i + 1; while (cw[j] == 0u) ++j;
             pn = (j << 5) + __builtin_ctz(cw[j]); }
      // curvature term (coords exact in f32: |values| < 2^9)
      const float Pr = (float)(p >> 8),  Pc = (float)(p & 255);
      const float Qr = (float)(pp >> 8), Qc = (float)(pp & 255);
      const float Rr = (float)(pn >> 8), Rc = (float)(pn & 255);
      const float v1r = Pr - Qr, v1c = Pc - Qc;
      const float v2r = Rr - Pr, v2c = Rc - Pc;
      const float cross = v1r * v2c - v1c * v2r;
      const float dotv  = v1r * v2r + v1c * v2c;
      const float n1 = sqrtf(v1r * v1r + v1c * v1c);
      const float n2 = sqrtf(v2r * v2r + v2c * v2c);
      float curv = 0.f;
      if (cross != 0.f) curv = 2.f * fabsf(cross) / (n1 * n2 + dotv);
      const float wgt = (cross >= 0.f) ? 1.f : MU_F;
      acc += wgt * curv * curv / (n1 + n2);
    }
  }

  // Block reduction (deterministic shape): wave32 shfl tree + 8 partials.
  for (int off = 16; off > 0; off >>= 1) acc += __shfl_xor(acc, off, 32);
  if ((threadIdx.x & 31) == 0) sPart[threadIdx.x >> 5] = acc;
  __syncthreads();
  if (threadIdx.x == 0) {
    float t = 0.f;
    for (int w = 0; w < 8; ++w) t += sPart[w];
    perImage[b] = t;
  }
}

// ---------------- Phase 3: deterministic final sum --------------------------
__global__ void final_sum_kernel(const float* __restrict__ perImage,
                                 float* __restrict__ out, int B, float invB) {
  if (threadIdx.x == 0 && blockIdx.x == 0) {
    float t = 0.f;
    for (int i = 0; i < B; ++i) t += perImage[i];
    out[0] = t * invB;                          // overwrites poisoned d_out
  }
}

extern "C" void kernel_launch(void* const* d_in, const int* in_sizes, int n_in,
                              void* d_out, int out_size, void* d_ws, size_t ws_size,
                              hipStream_t stream) {
  (void)n_in; (void)out_size; (void)ws_size;
  const float* target = (const float*)d_in[1];        // {"input","target"} order
  const int B = in_sizes[1] / (2 * PIX);              // 128
  unsigned* bits = (unsigned*)d_ws;                   // B * 8KB bitmasks (1 MB)
  float* perImage = (float*)((char*)d_ws + (size_t)B * WORDS * sizeof(unsigned));
  float* out = (float*)d_out;

  const int nWaves  = B * (WORDS / 8);                // one wave per 8 words
  const int wavesPerBlock = 256 / 32;
  const int blocks1 = (nWaves + wavesPerBlock - 1) / wavesPerBlock;

  hipLaunchKernelGGL(mask_bits_kernel, dim3(blocks1), dim3(256), 0, stream,
                     target, bits, nWaves);
  hipLaunchKernelGGL(contour_energy_kernel, dim3(B), dim3(256), 0, stream,
                     bits, perImage);
  hipLaunchKernelGGL(final_sum_kernel, dim3(1), dim3(32), 0, stream,
                     perImage, out, B, 1.0f / (float)B);
}